// EnhancedDynamicChannelAttention_38362647887961
// MI455X (gfx1250) — compile-verified
//
#include <hip/hip_runtime.h>

// EnhancedDynamicChannelAttention, algebraically exact rewrite for MI455X.
// Key identities (exact in real arithmetic, fp reassociation only):
//   scores[b,h,s] = (Wk^T q[b,h]) . f[b,s,h,:]   (+ const dropped: softmax-invariant)
//   ctx[b,h,:]    = Wv @ (sum_s attn_s f[b,s,h,:]) + bv   (since sum_s attn_s == 1)
// -> removes both big GEMMs (17.2 GFLOP), leaving a memory-bound kernel:
//    ~134 MB HBM read + ~134 MB HBM write (~11.5 us floor @ 23.3 TB/s);
//    intermediate passes are L2 hits (features 134 MB < 192 MB L2).
// The remaining dense GEMM (ctx = fbar * Wv^T, 128x128x128) runs on
// v_wmma_f32_16x16x4_f32 in full f32 precision.

#define Bsz 16
#define Ssz 2048
#define Csz 1024
#define Hsz 8
#define HDsz 128

typedef __attribute__((ext_vector_type(2))) float v2f;
typedef __attribute__((ext_vector_type(8))) float v8f;

// ---------------------------------------------------------------- kernel 1
// qk[r=b*H+h][d] = sum_e (pref[r]*Wq[e] + bq[e]) * Wk[e][d]
__global__ __launch_bounds__(HDsz) void qk_prep_kernel(
    const float* __restrict__ pref, const float* __restrict__ Wq,
    const float* __restrict__ bq, const float* __restrict__ Wk,
    float* __restrict__ qk) {
  const int r = blockIdx.x;      // 0..127  (b*H + h)
  const int d = threadIdx.x;     // 0..127
  const float p = pref[r];
  float acc = 0.0f;
  for (int e = 0; e < HDsz; ++e) {
    const float qe = p * Wq[e] + bq[e];
    acc = fmaf(qe, Wk[e * HDsz + d], acc);
  }
  qk[r * HDsz + d] = acc;
}

// ---------------------------------------------------------------- kernel 2
// scores[r][s] = qk[r,:] . f[b,s,h,:]   (one wave per s-row; lane = 4 floats)
__global__ __launch_bounds__(256) void scores_kernel(
    const float* __restrict__ feat, const float* __restrict__ qk,
    float* __restrict__ scores) {
  const int b = blockIdx.z, h = blockIdx.y;
  const int s0 = blockIdx.x * 64;
  const int lane = threadIdx.x & 31;
  const int wave = threadIdx.x >> 5;
  const int r = b * Hsz + h;
  const float4 qv = reinterpret_cast<const float4*>(qk + r * HDsz)[lane];
  const float* fb = feat + ((size_t)b * Ssz) * Csz + h * HDsz;
  for (int i = 0; i < 8; ++i) {
    const int s = s0 + wave * 8 + i;
    const float* rowp = fb + (size_t)s * Csz;
    if (i < 7) __builtin_prefetch(rowp + Csz, 0, 0);   // global_prefetch_b8
    const float4 f = reinterpret_cast<const float4*>(rowp)[lane];
    float v = f.x * qv.x + f.y * qv.y + f.z * qv.z + f.w * qv.w;
    for (int off = 16; off > 0; off >>= 1) v += __shfl_down(v, off, 32);
    if (lane == 0) scores[(size_t)r * Ssz + s] = v;
  }
}

// ---------------------------------------------------------------- kernel 3
// Per (b,h): softmax over S in LDS, then fbar[d] = sum_s attn_s * f[b,s,h,d]
__global__ __launch_bounds__(256) void softmax_fbar_kernel(
    const float* __restrict__ feat, const float* __restrict__ scores,
    float* __restrict__ fbar) {
  __shared__ float sc[Ssz];
  __shared__ float red[256];
  const int r = blockIdx.x;              // b*H + h
  const int b = r / Hsz, h = r % Hsz;
  const int t = threadIdx.x;             // 0..255
  const float* sr = scores + (size_t)r * Ssz;

  float m = -1e30f;
  for (int i = t; i < Ssz; i += 256) { const float x = sr[i]; sc[i] = x; m = fmaxf(m, x); }
  red[t] = m; __syncthreads();
  for (int o = 128; o > 0; o >>= 1) { if (t < o) red[t] = fmaxf(red[t], red[t + o]); __syncthreads(); }
  m = red[0]; __syncthreads();

  float ssum = 0.0f;
  for (int i = t; i < Ssz; i += 256) { const float e = __expf(sc[i] - m); sc[i] = e; ssum += e; }
  red[t] = ssum; __syncthreads();
  for (int o = 128; o > 0; o >>= 1) { if (t < o) red[t] += red[t + o]; __syncthreads(); }
  const float inv = 1.0f / red[0];
  __syncthreads();

  // weighted mean of feature rows (threads 0..127 own d; two s-stripes)
  const int d = t & (HDsz - 1);
  const int stripe = t >> 7;             // 0 or 1
  const float* fb = feat + ((size_t)b * Ssz) * Csz + h * HDsz + d;
  float acc = 0.0f;
  for (int s = stripe; s < Ssz; s += 2) {
    if (s + 16 < Ssz) __builtin_prefetch(fb + (size_t)(s + 16) * Csz, 0, 0);
    acc = fmaf(sc[s], fb[(size_t)s * Csz], acc);
  }
  red[t] = acc; __syncthreads();
  if (stripe == 0) fbar[(size_t)r * HDsz + d] = (acc + red[t + 128]) * inv;
}

// ---------------------------------------------------------------- kernel 4
// ctx = fbar(128x128) * Wv^T(128x128) + bv  via v_wmma_f32_16x16x4_f32.
// A 16x4 layout (ISA 7.12.2): lane<16 -> K = k0+{0,1} in regs {0,1};
// lane>=16 -> K = k0+{2,3}. B mirrored. C/D: VGPR i holds M = i + 8*(lane>=16).
__global__ __launch_bounds__(256) void ctx_wmma_kernel(
    const float* __restrict__ fbar, const float* __restrict__ Wv,
    const float* __restrict__ bv, float* __restrict__ ctx) {
  const int waveG = blockIdx.x * (blockDim.x >> 5) + (threadIdx.x >> 5);  // 0..63
  const int lane = threadIdx.x & 31;
  const int m0 = (waveG >> 3) * 16;
  const int n0 = (waveG & 7) * 16;
  const int half = lane >> 4;
  const int l16 = lane & 15;
  const float* arow = fbar + (m0 + l16) * HDsz;   // A row (M = m0+l16)
  const float* brow = Wv + (n0 + l16) * HDsz;     // B col n = Wv row (N = n0+l16)
  v8f c = {};
#pragma unroll
  for (int k0 = 0; k0 < HDsz; k0 += 4) {
    v2f a, bm;
    a.x  = arow[k0 + 2 * half];
    a.y  = arow[k0 + 2 * half + 1];
    bm.x = brow[k0 + 2 * half];
    bm.y = brow[k0 + 2 * half + 1];
    c = __builtin_amdgcn_wmma_f32_16x16x4_f32(
        /*neg_a=*/false, a, /*neg_b=*/false, bm,
        /*c_mod=*/(short)0, c, /*reuse_a=*/false, /*reuse_b=*/false);
  }
  const float bias = bv[n0 + l16];
#pragma unroll
  for (int i = 0; i < 8; ++i) {
    const int m = m0 + half * 8 + i;
    ctx[m * HDsz + n0 + l16] = c[i] + bias;
  }
}

// ---------------------------------------------------------------- kernel 5
// out[b,s,c] = features[b,s,c] + ctx[b, c]   (ctx is [B,H,HD] == [B,C] flat)
__global__ __launch_bounds__(256) void bcast_add_kernel(
    const float4* __restrict__ feat4, const float4* __restrict__ ctx4,
    float4* __restrict__ out4) {
  const size_t n4 = (size_t)Bsz * Ssz * Csz / 4;
  const size_t stride = (size_t)gridDim.x * blockDim.x;
  for (size_t i = (size_t)blockIdx.x * blockDim.x + threadIdx.x; i < n4; i += stride) {
    const size_t flat = i * 4;
    const int b = (int)(flat / ((size_t)Ssz * Csz));
    const int cidx = (int)(flat % Csz);
    const float4 f = feat4[i];
    const float4 cv = ctx4[(size_t)b * (Csz / 4) + (cidx >> 2)];
    float4 o;
    o.x = f.x + cv.x; o.y = f.y + cv.y; o.z = f.z + cv.z; o.w = f.w + cv.w;
    out4[i] = o;
  }
}

extern "C" void kernel_launch(void* const* d_in, const int* in_sizes, int n_in,
                              void* d_out, int out_size, void* d_ws, size_t ws_size,
                              hipStream_t stream) {
  (void)in_sizes; (void)n_in; (void)out_size; (void)ws_size;
  const float* features = (const float*)d_in[0];   // [B,S,C]
  const float* pref     = (const float*)d_in[1];   // [B,H]
  const float* Wq       = (const float*)d_in[2];   // [HD,1]
  const float* bq       = (const float*)d_in[3];   // [HD]
  const float* Wk       = (const float*)d_in[4];   // [HD,HD]
  // d_in[5] = bk: exactly cancelled (uniform softmax logit shift) -> unused.
  const float* Wv       = (const float*)d_in[6];   // [HD,HD]
  const float* bv       = (const float*)d_in[7];   // [HD]
  float* out = (float*)d_out;

  // Workspace layout (~1.2 MB):
  float* qk     = (float*)d_ws;                    // [B*H, HD]  64 KB
  float* fbar   = qk + Bsz * Hsz * HDsz;           // [B*H, HD]  64 KB
  float* ctx    = fbar + Bsz * Hsz * HDsz;         // [B*H, HD]  64 KB
  float* scores = ctx + Bsz * Hsz * HDsz;          // [B*H, S]    1 MB

  qk_prep_kernel<<<Bsz * Hsz, HDsz, 0, stream>>>(pref, Wq, bq, Wk, qk);

  dim3 sg(Ssz / 64, Hsz, Bsz);                     // 32 x 8 x 16 blocks
  scores_kernel<<<sg, 256, 0, stream>>>(features, qk, scores);

  softmax_fbar_kernel<<<Bsz * Hsz, 256, 0, stream>>>(features, scores, fbar);

  ctx_wmma_kernel<<<8, 256, 0, stream>>>(fbar, Wv, bv, ctx);   // 64 waves

  bcast_add_kernel<<<4096, 256, 0, stream>>>(
      (const float4*)features, (const float4*)ctx, (float4*)out);
}